// RasterPoints_11647951307198
// MI455X (gfx1250) — compile-verified
//
#include <hip/hip_runtime.h>

// Output: (B, T, H, W, P) float32, 80M elements = 320 MB. The job is a
// bandwidth-bound memset (13.7 us floor at 23.3 TB/s) plus a tiny scatter of
// 8000 ones. No matrix structure -> WMMA not applicable; optimize the CDNA5
// memory path instead: wave32 blocks, GLOBAL_STORE_B128 with NT temporal hint
// (output > 192 MB L2, so bypassing L2 retention is the right policy).

typedef __attribute__((ext_vector_type(4))) float floatx4;

__global__ void raster_zero_kernel(float* __restrict__ out, long long n) {
    const long long n4 = n >> 2;
    const long long tid = (long long)blockIdx.x * blockDim.x + threadIdx.x;
    const long long stride = (long long)gridDim.x * blockDim.x;
    floatx4* __restrict__ out4 = (floatx4*)out;
    floatx4 z;
    z.x = 0.0f; z.y = 0.0f; z.z = 0.0f; z.w = 0.0f;
    // Grid-stride fill with 16B non-temporal stores (global_store_b128 th:NT).
    for (long long j = tid; j < n4; j += stride) {
        __builtin_nontemporal_store(z, out4 + j);
    }
    // Tail (out_size is divisible by 4 here, but keep it correct in general).
    if (tid == 0) {
        for (long long j = (n4 << 2); j < n; ++j) out[j] = 0.0f;
    }
}

__global__ void raster_scatter_kernel(const float* __restrict__ x,
                                      const float* __restrict__ res,
                                      const float* __restrict__ org,
                                      float* __restrict__ out,
                                      int BT, int P, int H, int W) {
    const int i = blockIdx.x * blockDim.x + threadIdx.x;
    const int total = BT * P;
    if (i >= total) return;
    const int p  = i % P;
    const int bt = i / P;

    // x layout: [B*T, 2P] viewed as [B*T, P, 2] -> (x, y) pairs.
    const float px = x[(bt * P + p) * 2 + 0];
    const float py = x[(bt * P + p) * 2 + 1];
    const float r0 = res[bt * 2 + 0];
    const float r1 = res[bt * 2 + 1];
    const float o0 = org[bt * 2 + 0];
    const float o1 = org[bt * 2 + 1];

    // C int cast truncates toward zero, matching jnp astype(int32).
    const int row = (int)(py / r0 + o0);
    const int col = (int)(px / r1 + o1);

    if (row >= 0 && row < H && col >= 0 && col < W) {
        const long long idx =
            ((((long long)bt * H + row) * W + col) * (long long)P) + p;
        out[idx] = 1.0f;  // unique (b,t,p) slot -> no write conflicts
    }
}

extern "C" void kernel_launch(void* const* d_in, const int* in_sizes, int n_in,
                              void* d_out, int out_size, void* d_ws, size_t ws_size,
                              hipStream_t stream) {
    const float* x   = (const float*)d_in[0];  // [B, T, 2P]
    const float* res = (const float*)d_in[1];  // [B, T, 2]
    const float* org = (const float*)d_in[2];  // [B, T, 2]
    float* out = (float*)d_out;

    const int BT = in_sizes[1] / 2;            // B*T = 320
    const int P  = in_sizes[0] / (2 * BT);     // 25
    const int H = 100, W = 100;
    const long long n = (long long)out_size;   // 80,000,000

    // --- Pass 1: zero 320 MB with B128 NT stores -------------------------
    const int threads = 256;                   // 8 wave32 waves per block
    const long long n4 = n >> 2;
    long long want = (n4 + threads - 1) / threads;
    int zblocks = (want > 16384) ? 16384 : (int)want;  // ~5 float4/thread
    if (zblocks < 1) zblocks = 1;
    raster_zero_kernel<<<zblocks, threads, 0, stream>>>(out, n);

    // --- Pass 2: scatter 8000 ones (ordered after pass 1 by the stream) --
    const int total = BT * P;
    const int sblocks = (total + threads - 1) / threads;
    raster_scatter_kernel<<<sblocks, threads, 0, stream>>>(x, res, org, out,
                                                           BT, P, H, W);
}